// involution_49125835932189
// MI455X (gfx1250) — compile-verified
//
#include <hip/hip_runtime.h>
#include <stdint.h>

// ---------------------------------------------------------------------------
// Involution (B=8, C=512, 64x64, G=8, K=7, pad=3, stride=1), fp32.
// Memory-bound (185 MB unique @ 23.3 TB/s ~ 8 us; 1.64 GFLOP).
// CDNA5 data path: TDM tensor_load_to_lds for the weight band,
// global_load_async_to_lds_b128 for the x band (halo clipping), fp32 VALU FMAs.
// ---------------------------------------------------------------------------

#define Kk 7
#define TILE_H 8            // output rows per workgroup
#define CB 8                // channels per workgroup (of cpg=64)
#define SX_W 76             // 4 left pad + 64 + 8 right pad (odd-ish stride, 16B-aligned rows)
#define SX_H 14             // TILE_H + 6 halo rows
#define SX_FLOATS (CB * SX_H * SX_W)        // 8512
#define SW_PIX (TILE_H * 64)                // 512 pixels per tap
#define SW_FLOATS (49 * SW_PIX)             // 25088
#define LDS_BYTES ((SX_FLOATS + SW_FLOATS) * 4)   // 134400 B  (<320KB WGP LDS)

typedef uint32_t v4u __attribute__((ext_vector_type(4)));
typedef int32_t  v4i __attribute__((ext_vector_type(4)));
typedef int32_t  v8i __attribute__((ext_vector_type(8)));

typedef __attribute__((address_space(1))) v4i* g4_ptr;   // global int4*
typedef __attribute__((address_space(3))) v4i* l4_ptr;   // LDS int4*

__device__ __forceinline__ void async_ld_b128(const void* g, void* l) {
#if __has_builtin(__builtin_amdgcn_global_load_async_to_lds_b128)
  __builtin_amdgcn_global_load_async_to_lds_b128((g4_ptr)(v4i*)g, (l4_ptr)(v4i*)l, 0, 0);
#else
  asm volatile("global_load_async_to_lds_b128 %0, %1, off"
               :: "v"((uint32_t)(uintptr_t)l), "v"(g) : "memory");
#endif
}

__device__ __forceinline__ void wait_async0() {
#if __has_builtin(__builtin_amdgcn_s_wait_asynccnt)
  __builtin_amdgcn_s_wait_asynccnt(0);
#else
  asm volatile("s_wait_asynccnt 0x0" ::: "memory");
#endif
}

__device__ __forceinline__ void wait_tensor0() {
#if __has_builtin(__builtin_amdgcn_s_wait_tensorcnt)
  __builtin_amdgcn_s_wait_tensorcnt(0);
#else
  asm volatile("s_wait_tensorcnt 0x0" ::: "memory");
#endif
}

__global__ __launch_bounds__(256)
void involution_kernel(const float* __restrict__ x,
                       const float* __restrict__ w,
                       float* __restrict__ out) {
  extern __shared__ float smem[];
  float* sx = smem;               // [CB][SX_H][SX_W]
  float* sw = smem + SX_FLOATS;   // [49][SW_PIX]

  const int bid  = blockIdx.x;
  const int band = bid & 7;          // 8 bands of 8 rows
  const int cb   = (bid >> 3) & 7;   // 8 channel blocks of 8
  const int g    = (bid >> 6) & 7;
  const int b    = bid >> 9;
  const int ho0  = band * TILE_H;
  const int tid  = threadIdx.x;

  const int chan0 = (b * 512) + g * 64 + cb * CB;
  const float* xb = x + (size_t)chan0 * 4096;                  // x[b, chan0, 0, 0]
  const float* wb = w + ((size_t)(b * 8 + g) * 49) * 4096;     // weight[b,g,0,0,0,0]
  float*       ob = out + (size_t)chan0 * 4096;

  // ---- Phase 1a: TDM load of weight band: 49 taps x (8 rows * 64 cols) ----
  // 2-D tile: tile_dim0 = 512 contiguous floats (rows ho0..ho0+7 of a 64x64
  // plane are contiguous), tile_dim1 = 49, dim0 stride = 4096 floats.
#if __has_builtin(__builtin_amdgcn_tensor_load_to_lds)
  if (tid < 32) {
    const uint64_t gaddr = (uint64_t)(uintptr_t)(wb + ho0 * 64);
    const uint32_t laddr = (uint32_t)(uintptr_t)sw;   // LDS byte offset (low 32 bits of generic)
    v4u g0 = { 1u,                                    // count=1, user descriptor
               laddr,
               (uint32_t)(gaddr & 0xFFFFFFFFu),
               (uint32_t)((gaddr >> 32) & 0x01FFFFFFu) | 0x80000000u };  // type=2 ("image")
    v8i g1 = { (int)0x00020000,          // data_size=2 (4B), no mask/pad/iterate
               (int)(512u << 16),        // tensor_dim0[15:0]=512 in [31:16]
               (int)(49u  << 16),        // tensor_dim1[15:0]=49  in [31:16]
               (int)(512u << 16),        // tile_dim0=512 in [31:16]
               (int)49,                  // tile_dim1=49, tile_dim2=0
               (int)4096,                // tensor_dim0_stride = 4096 elems
               0, 0 };
    v4i g2 = { 0, 0, 0, 0 };
    v4i g3 = { 0, 0, 0, 0 };
#if defined(__clang_major__) && (__clang_major__ >= 23)
    v8i z8 = { 0, 0, 0, 0, 0, 0, 0, 0 };
    __builtin_amdgcn_tensor_load_to_lds(g0, g1, g2, g3, z8, 0);
#else
    __builtin_amdgcn_tensor_load_to_lds(g0, g1, g2, g3, 0);
#endif
    wait_tensor0();
  }
#else
  // Fallback: per-lane async loads for the weight band (49*512 floats).
  for (int i = tid; i < 49 * (SW_PIX / 4); i += 256) {
    const int k = i >> 7;          // /128 quads per tap
    const int q = i & 127;
    async_ld_b128(wb + (size_t)k * 4096 + ho0 * 64 + q * 4, &sw[k * SW_PIX + q * 4]);
  }
#endif

  // ---- Phase 1b: zero the pad columns of sx (disjoint from async targets) --
  {
    const float4 z4 = make_float4(0.f, 0.f, 0.f, 0.f);
    for (int i = tid; i < CB * SX_H * 3; i += 256) {
      const int row  = i / 3;           // 0..111  (c*14 + r)
      const int part = i % 3;
      const int col  = (part == 0) ? 0 : (64 + part * 4);   // 0, 68, 72
      *(float4*)&sx[row * SX_W + col] = z4;
    }
  }

  // ---- Phase 1c: async-load x band rows (clip halo rows to the image) -----
  {
    const float4 z4 = make_float4(0.f, 0.f, 0.f, 0.f);
    for (int i = tid; i < CB * SX_H * 16; i += 256) {
      const int q   = i & 15;           // float4 within row
      const int row = i >> 4;           // 0..111
      const int c   = row / SX_H;
      const int r   = row - c * SX_H;
      const int h   = ho0 - 3 + r;      // global image row, may be OOB
      float* lp = &sx[row * SX_W + 4 + q * 4];
      if ((unsigned)h < 64u) {
        async_ld_b128(xb + (size_t)c * 4096 + h * 64 + q * 4, lp);
      } else {
        *(float4*)lp = z4;              // padding rows -> zeros (DS store)
      }
    }
  }

  wait_async0();
  __syncthreads();

  // ---- Phase 2: compute. Each thread: 4 channels x 4 adjacent wo. ----------
  const int wo    = (tid & 15) * 4;       // 0..60
  const int ty    = (tid >> 4) & 7;       // 0..7
  const int c_sub = tid >> 7;             // 0..1
  const int pix   = ty * 64 + wo;

  float4 acc[4];
#pragma unroll
  for (int j = 0; j < 4; ++j) acc[j] = make_float4(0.f, 0.f, 0.f, 0.f);

#pragma unroll
  for (int kh = 0; kh < Kk; ++kh) {
    float4 wv[Kk];
#pragma unroll
    for (int kw = 0; kw < Kk; ++kw)
      wv[kw] = *(const float4*)&sw[(kh * Kk + kw) * SW_PIX + pix];

#pragma unroll
    for (int j = 0; j < 4; ++j) {
      const int c = c_sub + 2 * j;
      const float* rp = &sx[(c * SX_H + ty + kh) * SX_W + wo];
      const float4 a0 = *(const float4*)(rp);
      const float4 a1 = *(const float4*)(rp + 4);
      const float4 a2 = *(const float4*)(rp + 8);
      const float xr[12] = { a0.x, a0.y, a0.z, a0.w,
                             a1.x, a1.y, a1.z, a1.w,
                             a2.x, a2.y, a2.z, a2.w };
#pragma unroll
      for (int kw = 0; kw < Kk; ++kw) {
        acc[j].x = fmaf(xr[kw + 1], wv[kw].x, acc[j].x);
        acc[j].y = fmaf(xr[kw + 2], wv[kw].y, acc[j].y);
        acc[j].z = fmaf(xr[kw + 3], wv[kw].z, acc[j].z);
        acc[j].w = fmaf(xr[kw + 4], wv[kw].w, acc[j].w);
      }
    }
  }

#pragma unroll
  for (int j = 0; j < 4; ++j) {
    const int c = c_sub + 2 * j;
    *(float4*)(ob + (size_t)c * 4096 + (ho0 + ty) * 64 + wo) = acc[j];
  }
}

extern "C" void kernel_launch(void* const* d_in, const int* in_sizes, int n_in,
                              void* d_out, int out_size, void* d_ws, size_t ws_size,
                              hipStream_t stream) {
  (void)in_sizes; (void)n_in; (void)d_ws; (void)ws_size; (void)out_size;
  const float* x = (const float*)d_in[0];
  const float* w = (const float*)d_in[1];
  float* out = (float*)d_out;
  // grid: B(8) * G(8) * channel-blocks(8) * bands(8) = 4096 workgroups
  involution_kernel<<<4096, 256, LDS_BYTES, stream>>>(x, w, out);
}